// RangeMix_13486197310117
// MI455X (gfx1250) — compile-verified
//
#include <hip/hip_runtime.h>
#include <hip/hip_bf16.h>
#include <stdint.h>

// RangeMix for MI455X (gfx1250): pure bandwidth problem (~0.85 GB traffic, 0 FLOPs,
// ~37us HBM floor at 23.3 TB/s). CDNA5-specific path: async global<->LDS data mover
// (GLOBAL_LOAD_ASYNC_TO_LDS_B128 / GLOBAL_STORE_ASYNC_FROM_LDS_B128, ASYNCcnt) for
// the dominant embed traffic. Key optimization this round: each point's embed row is
// loaded from HBM exactly ONCE and fanned out to both output rows (i and N+i),
// halving the 256MB embed read stream (reuse distance exceeded L2 before).

#define B_  8
#define C_  5
#define H_  64
#define W_  2048
#define N_  1048576
#define HW_ (H_ * W_)

// Output layout (flat f32, reference return order)
#define OFF_SCAN   0                         // 5,242,880
#define OFF_COORD  5242880                   // 6,291,456
#define OFF_LABEL  11534336                  // 1,048,576
#define OFF_EMBED  12582912                  // 134,217,728
#define OFF_VALID  146800640                 // 2,097,152

__device__ __forceinline__ bool box_test(int b, int h, int w,
                                         const int* __restrict__ bw1,
                                         const int* __restrict__ bw2,
                                         const int* __restrict__ bh1,
                                         const int* __restrict__ bh2) {
    bool r = false;
#pragma unroll
    for (int f = 0; f < 4; ++f) {
        int o = b * 4 + f;
        r = r | ((h >= bh1[o]) & (h < bh2[o]) & (w >= bw1[o]) & (w < bw2[o]));
    }
    return r;
}

// ---- Kernel 1: mix_range_scan + mix_pred_label (one thread per (b,h,w)) ----
__global__ void k_scan_label(const float* __restrict__ scan,
                             const int*   __restrict__ pred,
                             const int*   __restrict__ u,
                             const int* __restrict__ bw1, const int* __restrict__ bw2,
                             const int* __restrict__ bh1, const int* __restrict__ bh2,
                             float* __restrict__ out_scan,
                             float* __restrict__ out_label) {
    int idx = blockIdx.x * blockDim.x + threadIdx.x;
    if (idx >= B_ * HW_) return;
    int w = idx & (W_ - 1);
    int h = (idx >> 11) & (H_ - 1);
    int b = idx >> 17;
    bool box = box_test(b, h, w, bw1, bw2, bh1, bh2);
    int sb = box ? u[b] : b;

    __builtin_nontemporal_store((float)pred[sb * HW_ + h * W_ + w], &out_label[idx]);
#pragma unroll
    for (int c = 0; c < C_; ++c) {
        float v = scan[(size_t)(sb * C_ + c) * HW_ + h * W_ + w];
        __builtin_nontemporal_store(v, &out_scan[(size_t)(b * C_ + c) * HW_ + h * W_ + w]);
    }
}

// ---- Kernel 2: mix_project_coord + valid (one thread per point) ----
__global__ void k_points(const int* __restrict__ pc,
                         const int* __restrict__ u,
                         const int* __restrict__ bw1, const int* __restrict__ bw2,
                         const int* __restrict__ bh1, const int* __restrict__ bh2,
                         float* __restrict__ out_coord,
                         float* __restrict__ out_valid) {
    int i = blockIdx.x * blockDim.x + threadIdx.x;
    if (i >= N_) return;
    int pb = pc[3 * i + 0];
    int pw = pc[3 * i + 1];
    int ph = pc[3 * i + 2];

    int cut = 0;                 // inv_perm[pb]: u is a permutation of 0..7
#pragma unroll
    for (int j = 0; j < B_; ++j)
        if (u[j] == pb) cut = j;

    bool v0 = !box_test(pb,  ph, pw, bw1, bw2, bh1, bh2);  // mix_keep
    bool v1 =  box_test(cut, ph, pw, bw1, bw2, bh1, bh2);  // in_box_cut

    size_t r0 = (size_t)3 * i;
    size_t r1 = (size_t)3 * ((size_t)N_ + i);
    __builtin_nontemporal_store(v0 ? (float)pb : -1.0f, &out_coord[r0 + 0]);
    __builtin_nontemporal_store(v0 ? (float)pw : -1.0f, &out_coord[r0 + 1]);
    __builtin_nontemporal_store(v0 ? (float)ph : -1.0f, &out_coord[r0 + 2]);
    __builtin_nontemporal_store(v1 ? (float)cut : -1.0f, &out_coord[r1 + 0]);
    __builtin_nontemporal_store(v1 ? (float)pw  : -1.0f, &out_coord[r1 + 1]);
    __builtin_nontemporal_store(v1 ? (float)ph  : -1.0f, &out_coord[r1 + 2]);

    __builtin_nontemporal_store(v0 ? 1.0f : 0.0f, &out_valid[i]);
    __builtin_nontemporal_store(v1 ? 1.0f : 0.0f, &out_valid[N_ + i]);
}

// ---- Kernel 3: mix_range_embed via async LDS data mover ----
// 16 threads per POINT (not per output row): each thread moves one 16B chunk of
// embed[i] into LDS once, then fans it out to BOTH output rows i and N+i with two
// async stores (LDS source address cndmask'd to a zeroed slot when that row is
// invalid). This reads the 256MB embed table exactly once.
__global__ void k_embed(const float4* __restrict__ embed,
                        const int*    __restrict__ pc,
                        const int*    __restrict__ u,
                        const int* __restrict__ bw1, const int* __restrict__ bw2,
                        const int* __restrict__ bh1, const int* __restrict__ bh2,
                        float4* __restrict__ out_embed) {
    __shared__ float4 stage[256];   // loaded embed chunks
    __shared__ float4 zbuf[256];    // per-thread zero chunks
    unsigned t = threadIdx.x;
    size_t gid = (size_t)blockIdx.x * 256 + t;
    int i = (int)(gid >> 4);          // point index in [0, N)
    int c = (int)(gid & 15);          // float4 chunk of the 64-float row
    if (i >= N_) return;

    int pb = pc[3 * i + 0];
    int pw = pc[3 * i + 1];
    int ph = pc[3 * i + 2];
    int cut = 0;
#pragma unroll
    for (int j = 0; j < B_; ++j)
        if (u[j] == pb) cut = j;

    bool v0 = !box_test(pb,  ph, pw, bw1, bw2, bh1, bh2);  // row i valid
    bool v1 =  box_test(cut, ph, pw, bw1, bw2, bh1, bh2);  // row N+i valid

    zbuf[t] = make_float4(0.0f, 0.0f, 0.0f, 0.0f);         // ds_store_b128
    unsigned stage_off = (unsigned)(size_t)(&stage[t]);    // LDS byte addr (low 32b of flat)
    unsigned z_off     = (unsigned)(size_t)(&zbuf[t]);

    if (v0 | v1) {
        unsigned long long src =
            (unsigned long long)(size_t)(embed + ((size_t)i * 16 + c));
        asm volatile("global_load_async_to_lds_b128 %0, %1, off"
                     :: "v"(stage_off), "v"(src) : "memory");
    }
    asm volatile("s_wait_asynccnt 0x0\n\ts_wait_dscnt 0x0" ::: "memory");

    unsigned s0 = v0 ? stage_off : z_off;
    unsigned s1 = v1 ? stage_off : z_off;
    unsigned long long d0 =
        (unsigned long long)(size_t)(out_embed + ((size_t)i * 16 + c));
    unsigned long long d1 =
        (unsigned long long)(size_t)(out_embed + ((size_t)(N_ + i) * 16 + c));

    asm volatile("global_store_async_from_lds_b128 %0, %1, off"
                 :: "v"(d0), "v"(s0) : "memory");
    asm volatile("global_store_async_from_lds_b128 %0, %1, off"
                 :: "v"(d1), "v"(s1) : "memory");
    // outstanding async stores drained by s_endpgm's implicit wait-idle
}

extern "C" void kernel_launch(void* const* d_in, const int* in_sizes, int n_in,
                              void* d_out, int out_size, void* d_ws, size_t ws_size,
                              hipStream_t stream) {
    const float* scan  = (const float*)d_in[0];
    const int*   pc    = (const int*)  d_in[1];
    const float* embed = (const float*)d_in[2];
    const int*   pred  = (const int*)  d_in[3];
    const int*   u     = (const int*)  d_in[4];
    const int*   bw1   = (const int*)  d_in[5];
    const int*   bw2   = (const int*)  d_in[6];
    const int*   bh1   = (const int*)  d_in[7];
    const int*   bh2   = (const int*)  d_in[8];

    float* out       = (float*)d_out;
    float* out_scan  = out + OFF_SCAN;
    float* out_coord = out + OFF_COORD;
    float* out_label = out + OFF_LABEL;
    float* out_embed = out + OFF_EMBED;
    float* out_valid = out + OFF_VALID;

    // (b,h,w) select: 2^20 threads
    k_scan_label<<<(B_ * HW_) / 256, 256, 0, stream>>>(
        scan, pred, u, bw1, bw2, bh1, bh2, out_scan, out_label);

    // per-point coords/valid: 2^20 threads
    k_points<<<N_ / 256, 256, 0, stream>>>(
        pc, u, bw1, bw2, bh1, bh2, out_coord, out_valid);

    // embed fan-out: N points * 16 chunks = 2^24 threads -> 65536 blocks
    k_embed<<<(unsigned)(((size_t)N_ * 16) / 256), 256, 0, stream>>>(
        (const float4*)embed, pc, u, bw1, bw2, bh1, bh2, (float4*)out_embed);
}